// TensorDecompositionLayer_85134841741546
// MI455X (gfx1250) — compile-verified
//
#include <hip/hip_runtime.h>
#include <cmath>

// ---------------------------------------------------------------------------
// Light-field Tucker-FISTA for MI455X (gfx1250), fp32 WMMA (V_WMMA_F32_16X16X4_F32)
// - B staged in LDS (zero-padded quad layout -> one ds_load_b64 per fragment)
// - two 16-row tiles per wave (each B fragment feeds two WMMAs)
// - A strength-reduced + one-quad software pipeline; vector A loads when legal
// ---------------------------------------------------------------------------

typedef float v2f __attribute__((ext_vector_type(2)));
typedef float v8f __attribute__((ext_vector_type(8)));

struct GemmP {
    const float* A;   // A operand (strided)
    const float* B;   // B operand (k-stride / n-stride)
    float*       C;   // output (or z buffer for FISTA epilogue)
    long long rows;   // GEMM rows
    int cols;         // GEMM cols
    int K;            // reduction length
    long long adiv, ahi, alo, aks;   // A offset = (row/adiv)*ahi + (row%adiv)*alo + k*aks
    long long bks, bns;              // B offset = k*bks + col*bns
    long long cdiv, chi, clo, cc;    // C offset (EPI==0)
    const float* xin;  // EPI==1: x for residual
    const float* scal; // EPI==2: scal[0]=L
    float*       y;    // EPI==2: y buffer (in/out)
    float        mom;  // EPI==2: momentum coefficient
};

__device__ __forceinline__ float softthr(float u, float t) {
    float m = fabsf(u) - t;
    m = m > 0.f ? m : 0.f;
    return u < 0.f ? -m : m;
}

// 8 waves/block; each wave owns TWO 16-row tiles x all NT column tiles.
// LDS B layout: Blds[q][col][kk] (kk=k%4), zero-padded in k and col -> the
// inner loop is branch- and select-free; each B fragment feeds 2 WMMAs.
template <int NT, int KMAX, int EPI, bool AVEC>
__global__ __launch_bounds__(256) void gemm_wmma(GemmP p) {
    constexpr int CP  = NT * 16;      // padded cols
    constexpr int QSZ = CP * 4;       // floats per k-quad
    __shared__ __align__(16) float Blds[(KMAX / 4) * QSZ];

    const int lane = threadIdx.x & 31;
    const int wid  = threadIdx.x >> 5;
    const int Kq    = (p.K + 3) >> 2;   // quads actually staged (<= KMAX/4)
    const int Kfull = p.K >> 2;         // guard-free quads

    // ---- cooperative, zero-padded B staging (before any early-out) ----
    for (int i = threadIdx.x; i < Kq * QSZ; i += 256) {
        const int q  = i / QSZ;
        const int r  = i - q * QSZ;
        const int c  = r >> 2;
        const int kk = r & 3;
        const int k  = q * 4 + kk;
        float val = 0.f;
        if (k < p.K && c < p.cols)
            val = p.B[(long long)k * p.bks + (long long)c * p.bns];
        Blds[i] = val;
    }
    __syncthreads();

    const long long rowTiles32 = (p.rows + 31) >> 5;
    const long long rt = (long long)blockIdx.x * 8 + wid;
    if (rt >= rowTiles32) return;          // uniform per wave (after barrier)
    const long long row0 = rt << 5;        // two 16-row tiles: row0, row0+16
    const int half = lane >> 4;            // 0: K=0,1   1: K=2,3  (16x4 f32 A layout)
    const int l15  = lane & 15;

    long long arow0 = row0 + l15;
    long long arow1 = row0 + 16 + l15;
    if (arow0 >= p.rows) arow0 = p.rows - 1;   // clamped; garbage masked at store
    if (arow1 >= p.rows) arow1 = p.rows - 1;
    const float* Ab0 = p.A + (arow0 / p.adiv) * p.ahi + (arow0 % p.adiv) * p.alo;
    const float* Ab1 = p.A + (arow1 / p.adiv) * p.ahi + (arow1 % p.adiv) * p.alo;

    v8f acc0[NT], acc1[NT];
    const v8f zv = {0.f, 0.f, 0.f, 0.f, 0.f, 0.f, 0.f, 0.f};
#pragma unroll
    for (int t = 0; t < NT; ++t) { acc0[t] = zv; acc1[t] = zv; }

    auto loadA = [&](const float* Ab, long long ko) -> v2f {
        if constexpr (AVEC) {              // aks==1, even row stride -> aligned b64
            return *(const v2f*)(Ab + ko);
        } else {
            v2f r;
            r.x = Ab[ko];
            r.y = Ab[ko + p.aks];
            return r;
        }
    };

    const float* blane = &Blds[half * 2 + l15 * 4];  // + q*QSZ + t*64
    long long koff = (long long)(half * 2) * p.aks;
    const long long kstep = 4 * p.aks;

    // ---- main loop: guard-free, A prefetched one quad ahead ----
    v2f a0c, a1c;
    if (Kfull > 0) { a0c = loadA(Ab0, koff); a1c = loadA(Ab1, koff); }
    for (int q = 0; q < Kfull; ++q) {
        v2f a0n = a0c, a1n = a1c;
        const long long koff2 = koff + kstep;
        if (q + 1 < Kfull) { a0n = loadA(Ab0, koff2); a1n = loadA(Ab1, koff2); }
        const float* bq = blane + (size_t)q * QSZ;
#pragma unroll
        for (int t = 0; t < NT; ++t) {
            const v2f bf = *(const v2f*)(bq + t * 64);
            acc0[t] = __builtin_amdgcn_wmma_f32_16x16x4_f32(
                false, a0c, false, bf, (short)0, acc0[t], false, false);
            acc1[t] = __builtin_amdgcn_wmma_f32_16x16x4_f32(
                false, a1c, false, bf, (short)0, acc1[t], false, false);
        }
        a0c = a0n;
        a1c = a1n;
        koff = koff2;
    }
    // ---- k tail (K%4 != 0): B padding is zero, A address clamped only here ----
    if (p.K & 3) {
        const int ka  = Kfull * 4 + half * 2;
        const int ka0 = ka       < p.K ? ka       : p.K - 1;
        const int ka1 = (ka + 1) < p.K ? (ka + 1) : p.K - 1;
        v2f a0, a1;
        a0.x = Ab0[(long long)ka0 * p.aks];
        a0.y = Ab0[(long long)ka1 * p.aks];
        a1.x = Ab1[(long long)ka0 * p.aks];
        a1.y = Ab1[(long long)ka1 * p.aks];
        const float* bq = blane + (size_t)Kfull * QSZ;
#pragma unroll
        for (int t = 0; t < NT; ++t) {
            const v2f bf = *(const v2f*)(bq + t * 64);
            acc0[t] = __builtin_amdgcn_wmma_f32_16x16x4_f32(
                false, a0, false, bf, (short)0, acc0[t], false, false);
            acc1[t] = __builtin_amdgcn_wmma_f32_16x16x4_f32(
                false, a1, false, bf, (short)0, acc1[t], false, false);
        }
    }

    // ---- epilogues ----
    float step = 0.f, thr = 0.f;
    if constexpr (EPI == 2) {
        float L = p.scal[0];
        L = L > 1e-6f ? L : 1e-6f;
        step = 1.f / L;
        thr  = 0.1f * step;   // COUPLE / L
    }

    auto epi = [&](v8f (&acc)[NT], long long trow0) {
#pragma unroll
        for (int t = 0; t < NT; ++t) {
            const int col = t * 16 + l15;
            if (col >= p.cols) continue;
#pragma unroll
            for (int j = 0; j < 8; ++j) {
                const long long row = trow0 + j + half * 8;   // C layout: VGPR j -> M=j / j+8
                if (row >= p.rows) continue;
                const float v = acc[t][j];
                if constexpr (EPI == 0) {
                    p.C[(row / p.cdiv) * p.chi + (row % p.cdiv) * p.clo + (long long)col * p.cc] = v;
                } else if constexpr (EPI == 1) {
                    // row=(p,h,st): resid[row*49+uv] = recon - x[(p,uv,st,h)]
                    const long long pp = row >> 9;
                    const int h  = (int)(row >> 6) & 7;
                    const int st = (int)row & 63;
                    p.C[row * 49 + col] = v - p.xin[((pp * 49 + col) * 64 + st) * 8 + h];
                } else if constexpr (EPI == 2) {
                    // row=(p,a,m), col=c : fused FISTA prox + momentum, in-place z/y
                    const long long idx = row * 16 + col;
                    const float u  = p.y[idx] - step * v;
                    const float zn = softthr(u, thr);
                    const float zo = p.C[idx];
                    p.C[idx] = zn;
                    p.y[idx] = zn + p.mom * (zn - zo);
                } else {
                    // EPI==3: final output permuted to (p,uv,st,h)
                    const long long pp = row >> 9;
                    const int h  = (int)(row >> 6) & 7;
                    const int st = (int)row & 63;
                    p.C[((pp * 49 + col) * 64 + st) * 8 + h] = v;
                }
            }
        }
    };
    epi(acc0, row0);
    epi(acc1, row0 + 16);
}

// --------------------------- small helper kernels ---------------------------

__global__ void fill_v0(float* v, int n) {
    int i = blockIdx.x * 256 + threadIdx.x;
    if (i < n) {
        unsigned u = (unsigned)i * 2654435761u;
        u ^= u >> 16; u *= 2246822519u; u ^= u >> 13; u *= 3266489917u; u ^= u >> 16;
        v[i] = ((float)(int)u) * (1.0f / 2147483648.0f);
    }
}

__global__ void sqsum_part(const float* w, int n, float* part) {
    __shared__ float sm[256];
    const int tid = threadIdx.x;
    float s = 0.f;
    for (int i = blockIdx.x * 256 + tid; i < n; i += 256 * 256) {
        float t = w[i];
        s += t * t;
    }
    sm[tid] = s;
    __syncthreads();
    for (int o = 128; o > 0; o >>= 1) {
        if (tid < o) sm[tid] += sm[tid + o];
        __syncthreads();
    }
    if (tid == 0) part[blockIdx.x] = sm[0];
}

__global__ void sqsum_fin(const float* part, float* scal) {
    __shared__ float sm[256];
    const int tid = threadIdx.x;
    sm[tid] = part[tid];
    __syncthreads();
    for (int o = 128; o > 0; o >>= 1) {
        if (tid < o) sm[tid] += sm[tid + o];
        __syncthreads();
    }
    if (tid == 0) {
        float nrm = sqrtf(sm[0]);
        scal[0] = nrm;                  // L after last iteration
        scal[1] = 1.f / (nrm + 1e-12f);
    }
}

__global__ void scale_vec(float* dst, const float* src, const float* scal, int n) {
    int i = blockIdx.x * 256 + threadIdx.x;
    if (i < n) dst[i] = src[i] * scal[1];
}

// ------------------------------- host side ----------------------------------

template <int NT, int KMAX, int EPI, bool AVEC>
static void run_gemm(const GemmP& p, hipStream_t s) {
    long long rowTiles32 = (p.rows + 31) >> 5;
    unsigned blocks = (unsigned)((rowTiles32 + 7) / 8);
    gemm_wmma<NT, KMAX, EPI, AVEC><<<dim3(blocks), dim3(256), 0, s>>>(p);
}

static const long long BIG = 1LL << 40;

static GemmP mk(const float* A, const float* B, float* C,
                long long rows, int cols, int K,
                long long adiv, long long ahi, long long alo, long long aks,
                long long bks, long long bns,
                long long cdiv, long long chi, long long clo, long long cc) {
    GemmP g{};
    g.A = A; g.B = B; g.C = C;
    g.rows = rows; g.cols = cols; g.K = K;
    g.adiv = adiv; g.ahi = ahi; g.alo = alo; g.aks = aks;
    g.bks = bks; g.bns = bns;
    g.cdiv = cdiv; g.chi = chi; g.clo = clo; g.cc = cc;
    return g;
}

extern "C" void kernel_launch(void* const* d_in, const int* in_sizes, int n_in,
                              void* d_out, int out_size, void* d_ws, size_t ws_size,
                              hipStream_t stream) {
    (void)in_sizes; (void)n_in; (void)out_size; (void)ws_size;
    const float* x  = (const float*)d_in[0];   // (P=256, uv=49, st=64, h=8)
    const float* da = (const float*)d_in[1];   // (98, 49)
    const float* ds = (const float*)d_in[2];   // (128, 64)
    const float* dc = (const float*)d_in[3];   // (16, 8)
    float* out = (float*)d_out;

    // ------- workspace carve-out (floats), 256-aligned -------
    float* wsf = (float*)d_ws;
    size_t off = 0;
    auto carve = [&](size_t n) { float* p = wsf + off; off += (n + 255) & ~(size_t)255; return p; };
    float* scal  = carve(16);         // [0]=L, [1]=1/nrm
    float* part  = carve(256);
    float* Ga    = carve(98 * 98);
    float* Gs    = carve(128 * 128);
    float* Gc    = carve(16 * 16);
    float* v     = carve(200704);     // (98,128,16)
    float* w     = carve(200704);
    float* zbuf  = carve(51380224);   // (P, 98, 128, 16)
    float* ybuf  = carve(51380224);
    float* buf1  = carve(25690112);   // t1 (p,a,m,h) / g2 (p,h,a,m)
    float* buf2  = carve(12845056);   // t2 (p,a,h,st) / g1 (p,h,st,a)
    float* resid = carve(6422528);    // (p,h,st,uv)

    hipMemsetAsync(zbuf, 0, (size_t)51380224 * sizeof(float), stream);
    hipMemsetAsync(ybuf, 0, (size_t)51380224 * sizeof(float), stream);

    // ------- Gram matrices: G = D * D^T -------
    run_gemm<7, 100, 0, false>(mk(da, da, Ga, 98, 98, 49,  BIG, 0, 49, 1,  1, 49,  BIG, 0, 98, 1), stream);
    run_gemm<8, 128, 0, true >(mk(ds, ds, Gs, 128, 128, 64, BIG, 0, 64, 1,  1, 64,  BIG, 0, 128, 1), stream);
    run_gemm<1, 16, 0, true >(mk(dc, dc, Gc, 16, 16, 8,   BIG, 0, 8, 1,   1, 8,   BIG, 0, 16, 1), stream);

    // ------- power iteration for L (layout-invariant eigenvalue) -------
    fill_v0<<<784, 256, 0, stream>>>(v, 200704);
    sqsum_part<<<256, 256, 0, stream>>>(v, 200704, part);
    sqsum_fin<<<1, 256, 0, stream>>>(part, scal);
    scale_vec<<<784, 256, 0, stream>>>(v, v, scal, 200704);

    for (int e = 0; e < 10; ++e) {
        // w1[a,m,d] = v[a,m,c] Gc[c,d]
        run_gemm<1, 16, 0, true >(mk(v, Gc, w, 12544, 16, 16,  BIG, 0, 16, 1,  16, 1,  BIG, 0, 16, 1), stream);
        // w2[a,d,n] = w1[a,m,d] Gs[m,n]   (rows=(a,d))
        run_gemm<8, 128, 0, false>(mk(w, Gs, v, 1568, 128, 128, 16, 2048, 1, 16, 128, 1, BIG, 0, 128, 1), stream);
        // v'[b,n,d] = w2[a,d,n] Ga[a,b]   (rows=(d,n), scatter back to (b,n,d))
        run_gemm<7, 100, 0, false>(mk(v, Ga, w, 2048, 98, 98,   BIG, 0, 1, 2048, 98, 1,  128, 1, 16, 2048), stream);
        sqsum_part<<<256, 256, 0, stream>>>(w, 200704, part);
        sqsum_fin<<<1, 256, 0, stream>>>(part, scal);
        scale_vec<<<784, 256, 0, stream>>>(v, w, scal, 200704);
    }
    // scal[0] now holds L.

    // ------- FISTA momentum coefficients (data-independent) -------
    float mom[15];
    {
        float t = 1.f;
        for (int i = 0; i < 15; ++i) {
            float tn = (1.f + sqrtf(1.f + 4.f * t * t)) * 0.5f;
            mom[i] = (t - 1.f) / tn;
            t = tn;
        }
    }

    // ------- FISTA main loop -------
    for (int it = 0; it < 15; ++it) {
        // t1[p,a,m,h] = y[p,a,m,c] dc[c,h]
        run_gemm<1, 16, 0, true >(mk(ybuf, dc, buf1, 3211264, 8, 16,  BIG, 0, 16, 1,  8, 1,  BIG, 0, 8, 1), stream);
        // t2[p,a,h,st] = t1[p,a,m,h] ds[m,st]   (rows=(p,a,h))
        run_gemm<4, 128, 0, false>(mk(buf1, ds, buf2, 200704, 64, 128, 8, 1024, 1, 8,  64, 1, BIG, 0, 64, 1), stream);
        // resid[p,h,st,uv] = t2[p,a,h,st] da[a,uv] - x   (rows=(p,h,st))
        {
            GemmP g = mk(buf2, da, resid, 131072, 49, 98,  512, 50176, 1, 512,  49, 1, 0, 0, 0, 0);
            g.xin = x;
            run_gemm<4, 100, 1, false>(g, stream);
        }
        // g1[p,h,st,a] = resid[p,h,st,uv] da^T[uv,a]
        run_gemm<7, 100, 0, false>(mk(resid, da, buf2, 131072, 98, 49, BIG, 0, 49, 1,  1, 49, BIG, 0, 98, 1), stream);
        // g2[p,h,a,m] = g1[p,h,st,a] ds^T[st,m]   (rows=(p,h,a))
        run_gemm<8, 128, 0, false>(mk(buf2, ds, buf1, 200704, 128, 64, 98, 6272, 1, 98, 1, 64, BIG, 0, 128, 1), stream);
        // g[p,a,m,c] = g2[p,h,a,m] dc^T[h,c]  + fused prox/momentum on z,y
        {
            GemmP g = mk(buf1, dc, zbuf, 3211264, 16, 8, 12544, 100352, 1, 12544, 1, 8, 0, 0, 0, 0);
            g.scal = scal;
            g.y = ybuf;
            g.mom = mom[it];
            run_gemm<1, 16, 2, false>(g, stream);
        }
    }

    // ------- final reconstruction from z, output permuted to (p,uv,st,h) -------
    run_gemm<1, 16, 0, true >(mk(zbuf, dc, buf1, 3211264, 8, 16,  BIG, 0, 16, 1,  8, 1,  BIG, 0, 8, 1), stream);
    run_gemm<4, 128, 0, false>(mk(buf1, ds, buf2, 200704, 64, 128, 8, 1024, 1, 8,  64, 1, BIG, 0, 64, 1), stream);
    {
        GemmP g = mk(buf2, da, out, 131072, 49, 98,  512, 50176, 1, 512, 49, 1, 0, 0, 0, 0);
        run_gemm<4, 100, 3, false>(g, stream);
    }
}